// GAT_23742579212603
// MI455X (gfx1250) — compile-verified
//
#include <hip/hip_runtime.h>
#include <math.h>

#define NNODES 50000
#define NEDGES 1600000
#define FDIM   128
#define NEG_SLOPE 0.2f

typedef __attribute__((ext_vector_type(2))) float v2f;
typedef __attribute__((ext_vector_type(8))) float v8f;

static inline int divUp(long long a, long long b) { return (int)((a + b - 1) / b); }

// ---------------------------------------------------------------------------
// GEMM: Out[NNODES, COLS] = X[NNODES, K] @ W[K, COLS], f32 WMMA 16x16x4.
// One wave per 16x16 output tile; K and COLS compile-time -> fully unrolled
// 32-WMMA chain with all VMEM addresses as base + immediate offset.
// ---------------------------------------------------------------------------
template <int K, int COLS>
__global__ void gat_gemm_wmma_f32(const float* __restrict__ X,
                                  const float* __restrict__ W,
                                  float* __restrict__ Out) {
    constexpr int colTiles = COLS / 16;
    constexpr int rowTiles = NNODES / 16;
    const int wave = (blockIdx.x * blockDim.x + threadIdx.x) >> 5;
    const int lane = threadIdx.x & 31;
    const int rowTile = wave / colTiles;
    const int colTile = wave - rowTile * colTiles;
    if (rowTile >= rowTiles) return;          // wave-uniform: EXEC all-ones inside

    const int m0 = rowTile << 4;
    const int n0 = colTile << 4;
    const int half = lane >> 4;               // 0: K=k0,k0+1   1: K=k0+2,k0+3
    const int l16  = lane & 15;

    v8f acc = {};
    const float* xr = X + (size_t)(m0 + l16) * K + 2 * half;     // A row (M = l16)
    const float* wc = W + (size_t)(2 * half) * COLS + n0 + l16;  // B col (N = l16)
    #pragma unroll
    for (int k0 = 0; k0 < K; k0 += 4) {
        v2f a, b;
        a.x = xr[k0];                                     // A K = k0+2*half
        a.y = xr[k0 + 1];                                 // A K = k0+2*half+1
        b.x = wc[k0 * COLS];                              // B K = k0+2*half
        b.y = wc[k0 * COLS + COLS];                       // B K = k0+2*half+1
        acc = __builtin_amdgcn_wmma_f32_16x16x4_f32(
            false, a, false, b, (short)0, acc, false, false);
    }
    // C/D layout: VGPR i holds M = i + 8*half, N = l16
    float* orow = Out + (size_t)(m0 + 8 * half) * COLS + n0 + l16;
    #pragma unroll
    for (int i = 0; i < 8; ++i) {
        orow[i * COLS] = acc[i];
    }
}

// ---------------------------------------------------------------------------
// Vector helpers (16B-aligned contiguous loads)
// ---------------------------------------------------------------------------
template <int H>
__device__ __forceinline__ void loadH(const float* __restrict__ p, float* out) {
    if constexpr (H % 4 == 0) {
        #pragma unroll
        for (int i = 0; i < H / 4; ++i) {
            float4 v = reinterpret_cast<const float4*>(p)[i];
            out[4 * i + 0] = v.x; out[4 * i + 1] = v.y;
            out[4 * i + 2] = v.z; out[4 * i + 3] = v.w;
        }
    } else {
        #pragma unroll
        for (int i = 0; i < H; ++i) out[i] = p[i];
    }
}

__device__ __forceinline__ float leaky(float v) {
    return v >= 0.f ? v : NEG_SLOPE * v;
}

__device__ __forceinline__ void atomicMaxF(float* addr, float val) {
    if (val >= 0.f)
        atomicMax((int*)addr, __float_as_int(val));
    else
        atomicMin((unsigned int*)addr, __float_as_uint(val));
}

// ---------------------------------------------------------------------------
// el[n,h] = sum_d F[n,h,d]*al[h,d] ; er likewise with ar.   (thread per n*H)
// ---------------------------------------------------------------------------
template <int H, int D>
__global__ void gat_attn_scores(const float* __restrict__ F,
                                const float* __restrict__ al,
                                const float* __restrict__ ar,
                                float* __restrict__ el,
                                float* __restrict__ er) {
    int idx = blockIdx.x * blockDim.x + threadIdx.x;
    if (idx >= NNODES * H) return;
    int h = idx % H;
    int n = idx / H;
    const float* f = F + (size_t)n * H * D + h * D;
    const float* a = al + h * D;
    const float* b = ar + h * D;
    float sl = 0.f, sr = 0.f;
    #pragma unroll
    for (int d = 0; d < D; d += 4) {
        float4 fv = *reinterpret_cast<const float4*>(f + d);
        float4 av = *reinterpret_cast<const float4*>(a + d);
        float4 bv = *reinterpret_cast<const float4*>(b + d);
        sl += fv.x * av.x + fv.y * av.y + fv.z * av.z + fv.w * av.w;
        sr += fv.x * bv.x + fv.y * bv.y + fv.z * bv.z + fv.w * bv.w;
    }
    el[idx] = sl;
    er[idx] = sr;
}

// ---------------------------------------------------------------------------
// Init: m = -inf, s = 0 (N*H entries), rst = 0 (N*H*D entries)
// ---------------------------------------------------------------------------
__global__ void gat_init(float* __restrict__ m, float* __restrict__ s,
                         float* __restrict__ rst, int nH, int nHD) {
    int idx = blockIdx.x * blockDim.x + threadIdx.x;
    if (idx < nHD) rst[idx] = 0.f;
    if (idx < nH) {
        m[idx] = -INFINITY;
        s[idx] = 0.f;
    }
}

// ---------------------------------------------------------------------------
// Pass 1: segment max of leaky(el[src]+er[dst]) over dst.   (thread per edge)
// ---------------------------------------------------------------------------
template <int H>
__global__ void gat_edge_max(const int* __restrict__ src, const int* __restrict__ dst,
                             const float* __restrict__ el, const float* __restrict__ er,
                             float* __restrict__ m) {
    int e = blockIdx.x * blockDim.x + threadIdx.x;
    if (e >= NEDGES) return;
    int s = src[e], d = dst[e];
    float ev[H], rv[H];
    loadH<H>(el + (size_t)s * H, ev);
    loadH<H>(er + (size_t)d * H, rv);
    #pragma unroll
    for (int h = 0; h < H; ++h)
        atomicMaxF(&m[(size_t)d * H + h], leaky(ev[h] + rv[h]));
}

// ---------------------------------------------------------------------------
// Pass 2: s[dst,h] += exp(e - m[dst,h])                     (thread per edge)
// ---------------------------------------------------------------------------
template <int H>
__global__ void gat_edge_expsum(const int* __restrict__ src, const int* __restrict__ dst,
                                const float* __restrict__ el, const float* __restrict__ er,
                                const float* __restrict__ m, float* __restrict__ ssum) {
    int e = blockIdx.x * blockDim.x + threadIdx.x;
    if (e >= NEDGES) return;
    int s = src[e], d = dst[e];
    float ev[H], rv[H], mv[H];
    loadH<H>(el + (size_t)s * H, ev);
    loadH<H>(er + (size_t)d * H, rv);
    loadH<H>(m  + (size_t)d * H, mv);
    #pragma unroll
    for (int h = 0; h < H; ++h) {
        float ex = expf(leaky(ev[h] + rv[h]) - mv[h]);
        atomicAdd(&ssum[(size_t)d * H + h], ex);
    }
}

// ---------------------------------------------------------------------------
// Pass 3: rst[dst,h,dd] += F[src,h,dd] * alpha(e,h)   (thread per edge*HD/4)
// ---------------------------------------------------------------------------
template <int H, int D>
__global__ void gat_edge_aggregate(const int* __restrict__ src, const int* __restrict__ dst,
                                   const float* __restrict__ el, const float* __restrict__ er,
                                   const float* __restrict__ m, const float* __restrict__ ssum,
                                   const float* __restrict__ F, float* __restrict__ rst) {
    constexpr int HD = H * D;
    constexpr int CHUNKS = HD / 4;          // float4 per thread
    int idx = blockIdx.x * blockDim.x + threadIdx.x;
    if (idx >= NEDGES * CHUNKS) return;
    int e  = idx / CHUNKS;
    int q  = idx - e * CHUNKS;
    int c0 = q * 4;
    int h  = c0 / D;
    int s = src[e], d = dst[e];
    float v = leaky(el[(size_t)s * H + h] + er[(size_t)d * H + h]);
    float alpha = expf(v - m[(size_t)d * H + h]) / ssum[(size_t)d * H + h];
    float4 f = *reinterpret_cast<const float4*>(F + (size_t)s * HD + c0);
    float* r = rst + (size_t)d * HD + c0;
    atomicAdd(r + 0, f.x * alpha);
    atomicAdd(r + 1, f.y * alpha);
    atomicAdd(r + 2, f.z * alpha);
    atomicAdd(r + 3, f.w * alpha);
}

// ---------------------------------------------------------------------------
// Finalize: out = act( rst + res + bias )                   (thread per N*HD)
// ---------------------------------------------------------------------------
__global__ void gat_finalize(const float* __restrict__ rst, const float* __restrict__ res,
                             const float* __restrict__ bias, float* __restrict__ out,
                             int HD, int act) {
    int idx = blockIdx.x * blockDim.x + threadIdx.x;
    if (idx >= NNODES * HD) return;
    int c = idx % HD;
    float v = rst[idx] + bias[c];
    if (res) v += res[idx];
    if (act) v = (v > 0.f) ? v : expm1f(v);
    out[idx] = v;
}

// ---------------------------------------------------------------------------
// Host-side: enqueue one GAT layer (all kernels stream-ordered)
// ---------------------------------------------------------------------------
template <int H, int D>
static void run_gat_layer(const float* hin, const float* W, const float* al,
                          const float* ar, const float* bias, const float* res,
                          bool act, const int* src, const int* dst,
                          float* F, float* EL, float* ER, float* Mb, float* Sb,
                          float* RST, float* hout, hipStream_t stream) {
    constexpr int HD = H * D;
    // projection: F = hin @ W  ([N,128] @ [128,HD])
    {
        long long waves = (long long)(NNODES / 16) * (HD / 16);
        int blocks = divUp(waves * 32, 256);
        gat_gemm_wmma_f32<FDIM, HD><<<blocks, 256, 0, stream>>>(hin, W, F);
    }
    gat_attn_scores<H, D><<<divUp((long long)NNODES * H, 256), 256, 0, stream>>>(F, al, ar, EL, ER);
    gat_init<<<divUp((long long)NNODES * HD, 256), 256, 0, stream>>>(Mb, Sb, RST, NNODES * H, NNODES * HD);
    gat_edge_max<H><<<divUp(NEDGES, 256), 256, 0, stream>>>(src, dst, EL, ER, Mb);
    gat_edge_expsum<H><<<divUp(NEDGES, 256), 256, 0, stream>>>(src, dst, EL, ER, Mb, Sb);
    gat_edge_aggregate<H, D><<<divUp((long long)NEDGES * (HD / 4), 256), 256, 0, stream>>>(
        src, dst, EL, ER, Mb, Sb, F, RST);
    gat_finalize<<<divUp((long long)NNODES * HD, 256), 256, 0, stream>>>(RST, res, bias, hout, HD, act ? 1 : 0);
}

extern "C" void kernel_launch(void* const* d_in, const int* in_sizes, int n_in,
                              void* d_out, int out_size, void* d_ws, size_t ws_size,
                              hipStream_t stream) {
    const float* x    = (const float*)d_in[0];
    const int*   src  = (const int*)d_in[1];
    const int*   dst  = (const int*)d_in[2];
    const float* W0   = (const float*)d_in[3];
    const float* al0  = (const float*)d_in[4];
    const float* ar0  = (const float*)d_in[5];
    const float* b0   = (const float*)d_in[6];
    const float* W1   = (const float*)d_in[7];
    const float* al1  = (const float*)d_in[8];
    const float* ar1  = (const float*)d_in[9];
    const float* b1   = (const float*)d_in[10];
    const float* W2   = (const float*)d_in[11];
    const float* al2  = (const float*)d_in[12];
    const float* ar2  = (const float*)d_in[13];
    const float* b2   = (const float*)d_in[14];
    const float* resW2= (const float*)d_in[15];

    const size_t N128 = (size_t)NNODES * FDIM;
    const size_t NH   = (size_t)NNODES * 8;
    float* F   = (float*)d_ws;     // [N,128]
    float* RST = F   + N128;       // [N,128]
    float* H0  = RST + N128;       // [N,128]  (layer-2: reused as res2 [N,64])
    float* H1  = H0  + N128;       // [N,128]
    float* EL  = H1  + N128;       // [N,8]
    float* ER  = EL  + NH;         // [N,8]
    float* Mb  = ER  + NH;         // [N,8]
    float* Sb  = Mb  + NH;         // [N,8]

    // ---- layer 0: x -> H0, 8 heads x 16, no residual, elu ----
    run_gat_layer<8, 16>(x, W0, al0, ar0, b0, /*res=*/nullptr, /*act=*/true,
                         src, dst, F, EL, ER, Mb, Sb, RST, H0, stream);

    // ---- layer 1: H0 -> H1, 8 heads x 16, identity residual, elu ----
    run_gat_layer<8, 16>(H0, W1, al1, ar1, b1, /*res=*/H0, /*act=*/true,
                         src, dst, F, EL, ER, Mb, Sb, RST, H1, stream);

    // ---- layer 2: H1 -> out, 1 head x 64, linear residual, no act ----
    float* RES2 = H0;  // H0 dead; reuse first N*64 floats
    {
        long long waves = (long long)(NNODES / 16) * (64 / 16);
        int blocks = divUp(waves * 32, 256);
        gat_gemm_wmma_f32<FDIM, 64><<<blocks, 256, 0, stream>>>(H1, resW2, RES2);
    }
    run_gat_layer<1, 64>(H1, W2, al2, ar2, b2, /*res=*/RES2, /*act=*/false,
                         src, dst, F, EL, ER, Mb, Sb, RST, (float*)d_out, stream);
}